// LiftSplatShoot_3272765080377
// MI455X (gfx1250) — compile-verified
//
#include <hip/hip_runtime.h>

typedef __attribute__((ext_vector_type(2))) float v2f;
typedef __attribute__((ext_vector_type(8))) float v8f;

#define NB    8
#define NCAM  6
#define BN    48          // NB*NCAM
#define CCH   64
#define DD    41
#define DP    48          // D padded to 3 full 16-tiles
#define FHH   16
#define FWW   44
#define NX    200
#define PLANE (NX*NX)

// ---------------------------------------------------------------- zero fill
__global__ void zero_f4(float4* __restrict__ p, long n4) {
    long i = (long)blockIdx.x * blockDim.x + threadIdx.x;
    long st = (long)gridDim.x * blockDim.x;
    float4 z = {0.f, 0.f, 0.f, 0.f};
    for (; i < n4; i += st) p[i] = z;
}

// ---------------------------------------------------------------- lift+splat
// One 256-thread block (8 waves) per (bn, fh) image row.
// LDS layouts are pixel-major and D-padded to 48 so the hot loop has no
// d-bounds branches: s_depth[fw][48] (pad rows = 0), s_feat[fw][64],
// s_seg[fw][48] (pad rows = -1, so culling == padding).
// Each wave owns pixels fw = wave, wave+8, ...; per pixel it loads
// 3 A-fragments, 4 B-fragments and 6 seg-int4s once, then fires 12
// v_wmma_f32_16x16x4_f32 rank-1 outer products and scatters with f32 atomics.
template <int USE_WS>
__global__ void lss_scatter(const float* __restrict__ dl,
                            const float* __restrict__ feat,
                            const float* __restrict__ geom,
                            float* __restrict__ acc) {
    __shared__ float s_depth[FWW * DP];
    __shared__ float s_feat[FWW * CCH];
    alignas(16) __shared__ int s_seg[FWW * DP];

    const int bn  = blockIdx.x / FHH;
    const int fh  = blockIdx.x % FHH;
    const int b   = bn / NCAM;
    const int tid = threadIdx.x;

    // ---- stage depth logits (global-coalesced read, transposed LDS write)
    for (int i = tid; i < DD * FWW; i += 256) {
        int d = i / FWW, fw = i % FWW;
        s_depth[fw * DP + d] = dl[((bn * DD + d) * FHH + fh) * FWW + fw];
    }
    // pad rows d=41..47: depth=0, seg=-1
    for (int i = tid; i < FWW * (DP - DD); i += 256) {
        int fw = i / (DP - DD), d = DD + i % (DP - DD);
        s_depth[fw * DP + d] = 0.f;
        s_seg[fw * DP + d]   = -1;
    }
    __syncthreads();

    // ---- per-pixel softmax over D (one thread per fw column)
    if (tid < FWW) {
        float* col = &s_depth[tid * DP];
        float m = -3.4e38f;
        for (int d = 0; d < DD; ++d) m = fmaxf(m, col[d]);
        float s = 0.f;
        for (int d = 0; d < DD; ++d) {
            float e = expf(col[d] - m);
            col[d] = e;
            s += e;
        }
        float r = 1.f / s;
        for (int d = 0; d < DD; ++d) col[d] *= r;
    }

    // ---- stage features (coalesced read, transposed LDS write)
    for (int i = tid; i < CCH * FWW; i += 256) {
        int c = i / FWW, fw = i % FWW;
        s_feat[fw * CCH + c] = feat[((bn * CCH + c) * FHH + fh) * FWW + fw];
    }

    // ---- voxel segment per point
    for (int i = tid; i < DD * FWW; i += 256) {
        int d = i / FWW, fw = i % FWW;
        long gb = 3L * (((long)(bn * DD + d) * FHH + fh) * FWW + fw);
        float gx = geom[gb + 0], gy = geom[gb + 1], gz = geom[gb + 2];
        int xi = (int)floorf((gx - (-50.0f)) / 0.5f);
        int yi = (int)floorf((gy - (-50.0f)) / 0.5f);
        int zi = (int)floorf((gz - (-10.0f)) / 20.0f);
        bool kept = (xi >= 0) & (xi < NX) & (yi >= 0) & (yi < NX) & (zi == 0);
        s_seg[fw * DP + d] = kept ? (xi * NX + yi) : -1;
    }
    __syncthreads();

    const int wave = tid >> 5;
    const int lane = tid & 31;
    const int nlo  = lane & 15;        // N index within 16-wide tile
    const int hi   = lane >> 4;        // C/D row set: M = v + 8*hi

    const long wsBase = (long)b * PLANE * CCH;   // USE_WS accumulator base

    for (int fw = wave; fw < FWW; fw += 8) {
        const float* dp = &s_depth[fw * DP];
        const float* fp = &s_feat[fw * CCH];
        const int*   sp = &s_seg[fw * DP];

        // A fragments (depth, K=0 slot only) and B fragments (feat)
        float aarr[3], barr[4];
#pragma unroll
        for (int dt = 0; dt < 3; ++dt) {
            float v = dp[dt * 16 + nlo];
            aarr[dt] = hi ? 0.f : v;
        }
#pragma unroll
        for (int ct = 0; ct < 4; ++ct) {
            float v = fp[ct * 16 + nlo];
            barr[ct] = hi ? 0.f : v;
        }

        // seg values: each lane needs 8 contiguous ints -> two ds_load_b128
        int segv[3][8];
#pragma unroll
        for (int dt = 0; dt < 3; ++dt) {
            const int4* q = reinterpret_cast<const int4*>(sp + dt * 16 + hi * 8);
            int4 q0 = q[0], q1 = q[1];
            segv[dt][0] = q0.x; segv[dt][1] = q0.y;
            segv[dt][2] = q0.z; segv[dt][3] = q0.w;
            segv[dt][4] = q1.x; segv[dt][5] = q1.y;
            segv[dt][6] = q1.z; segv[dt][7] = q1.w;
        }

#pragma unroll
        for (int dt = 0; dt < 3; ++dt) {
            v2f a; a.x = aarr[dt]; a.y = 0.f;
#pragma unroll
            for (int ct = 0; ct < 4; ++ct) {
                v2f bm; bm.x = barr[ct]; bm.y = 0.f;
                v8f cacc = {0.f, 0.f, 0.f, 0.f, 0.f, 0.f, 0.f, 0.f};
                cacc = __builtin_amdgcn_wmma_f32_16x16x4_f32(
                    false, a, false, bm, (short)0, cacc, false, false);

                const int cch = ct * 16 + nlo;
#pragma unroll
                for (int v = 0; v < 8; ++v) {
                    int seg = segv[dt][v];
                    if (seg >= 0) {
                        if (USE_WS) {
                            // channel-contiguous accumulator: [b, xi, yi, c]
                            atomicAdd(acc + wsBase + (long)seg * CCH + cch,
                                      cacc[v]);
                        } else {
                            // direct reference layout: [b, c, xi, yi]
                            atomicAdd(acc + (long)(b * CCH + cch) * PLANE + seg,
                                      cacc[v]);
                        }
                    }
                }
            }
        }
    }
}

// ---------------------------------------------------------------- transpose
// ws [b, xi, yi, c] -> out [b, c, xi, yi]; one block per (b, xi) row,
// LDS-tiled with +1 padding so both sides stay bank-conflict free.
__global__ void lss_transpose(const float* __restrict__ ws,
                              float* __restrict__ out) {
    __shared__ float s[NX * (CCH + 1)];
    const int bx = blockIdx.x;
    const int b  = bx / NX;
    const int xi = bx % NX;
    const float* src = ws + ((long)b * PLANE + (long)xi * NX) * CCH;
    for (int i = threadIdx.x; i < NX * CCH; i += blockDim.x) {
        int yi = i / CCH, c = i % CCH;
        s[yi * (CCH + 1) + c] = src[i];            // fully coalesced read
    }
    __syncthreads();
    for (int i = threadIdx.x; i < NX * CCH; i += blockDim.x) {
        int c = i / NX, yi = i % NX;
        out[((long)(b * CCH + c) * NX + xi) * NX + yi] = s[yi * (CCH + 1) + c];
    }
}

// ---------------------------------------------------------------- launch
extern "C" void kernel_launch(void* const* d_in, const int* in_sizes, int n_in,
                              void* d_out, int out_size, void* d_ws, size_t ws_size,
                              hipStream_t stream) {
    const float* dl   = (const float*)d_in[0];  // depth_logits [48,41,16,44]
    const float* feat = (const float*)d_in[1];  // feat         [48,64,16,44]
    const float* geom = (const float*)d_in[2];  // geom         [8,6,41,16,44,3]
    float* out = (float*)d_out;                 // [8, 64, 200, 200]

    const size_t outElems = (size_t)NB * CCH * PLANE;   // 20,480,000
    const size_t wsNeed   = outElems * sizeof(float);   // ~82 MB

    if (ws_size >= wsNeed) {
        float* ws = (float*)d_ws;
        zero_f4<<<2048, 256, 0, stream>>>((float4*)ws, (long)(outElems / 4));
        lss_scatter<1><<<BN * FHH, 256, 0, stream>>>(dl, feat, geom, ws);
        lss_transpose<<<NB * NX, 256, 0, stream>>>(ws, out);
    } else {
        zero_f4<<<2048, 256, 0, stream>>>((float4*)out, (long)(outElems / 4));
        lss_scatter<0><<<BN * FHH, 256, 0, stream>>>(dl, feat, geom, out);
    }
}